// NLFIter_15796889715112
// MI455X (gfx1250) — compile-verified
//
#include <hip/hip_runtime.h>
#include <stdint.h>

// Problem constants (fixed by the reference setup).
#define B_ 4
#define C_ 32
#define H_ 256
#define W_ 512

// W-sweep tiling.
#define TW 16                 // columns per chunk
#define PITCH 20              // floats per tile row (16 data + 4 pad) -> 80B, 16B aligned, bank-friendly
#define NCHUNK (W_ / TW)
#define TILEF (H_ * PITCH)    // floats per tile (one tensor plane chunk)

// ---------------------------------------------------------------------------
// CDNA5 async global->LDS helpers (inline asm; tracked by ASYNCcnt).
// ---------------------------------------------------------------------------
__device__ __forceinline__ uint32_t lds_off32(const void* p) {
  // Generic pointer to LDS: low 32 bits are the LDS byte address.
  return (uint32_t)(uintptr_t)p;
}

__device__ __forceinline__ void async_load_b128(uint32_t lds, uint64_t gaddr) {
  // VDST = LDS byte address (VGPR), VADDR = 64-bit global address, no SADDR.
  asm volatile("global_load_async_to_lds_b128 %0, %1, off"
               :: "v"(lds), "v"(gaddr)
               : "memory");
}

__device__ __forceinline__ void wait_async_all() {
  asm volatile("s_wait_asynccnt 0x0" ::: "memory");
}
__device__ __forceinline__ void wait_async_20() {
  // Each thread issues 20 async loads per chunk; <=20 outstanding means the
  // previous chunk's 20 (loads complete in order) are done.
  asm volatile("s_wait_asynccnt 0x14" ::: "memory");
}

// ---------------------------------------------------------------------------
// H sweep (DIR=+1 down, DIR=-1 up). One block per (b,c) plane.
// 128 threads, each owns 4 consecutive columns (float4 loads/stores).
// Double-buffered LDS previous-row -> one barrier per row. Next row's x/g are
// prefetched into registers so global latency hides behind the barrier.
// Safe in-place (in may == out).
// ---------------------------------------------------------------------------
template <int DIR>
__launch_bounds__(W_ / 4)
__global__ void nlf_sweepH(const float* in, const float* __restrict__ g, float* out) {
  __shared__ float prevbuf[2][W_];
  const int t  = threadIdx.x;       // 0..127
  const int w0 = t * 4;
  const int bc = blockIdx.x;        // b*C + c
  const int b  = bc >> 5;           // / C_
  const float* xin = in  + (size_t)bc * (H_ * W_);
  float*       xo  = out + (size_t)bc * (H_ * W_);
  const float* gb  = g   + (size_t)b  * (4 * H_ * W_);

  prevbuf[0][w0 + 0] = 0.f; prevbuf[0][w0 + 1] = 0.f;
  prevbuf[0][w0 + 2] = 0.f; prevbuf[0][w0 + 3] = 0.f;

  int h = (DIR > 0) ? 0 : (H_ - 1);
  float4 xv = *(const float4*)&xin[(size_t)h * W_ + w0];
  float4 a0 = *(const float4*)&gb[(size_t)(0 * H_ + h) * W_ + w0];
  float4 a1 = *(const float4*)&gb[(size_t)(1 * H_ + h) * W_ + w0];
  float4 a2 = *(const float4*)&gb[(size_t)(2 * H_ + h) * W_ + w0];
  float4 a3 = *(const float4*)&gb[(size_t)(3 * H_ + h) * W_ + w0];
  __syncthreads();

  for (int i = 0; i < H_; ++i) {
    const int hn = (DIR > 0) ? (i + 1) : (H_ - 2 - i);
    float4 nx = xv, n0 = a0, n1 = a1, n2 = a2, n3 = a3;
    if (i + 1 < H_) {  // prefetch next row (overlaps with barrier + LDS below)
      nx = *(const float4*)&xin[(size_t)hn * W_ + w0];
      n0 = *(const float4*)&gb[(size_t)(0 * H_ + hn) * W_ + w0];
      n1 = *(const float4*)&gb[(size_t)(1 * H_ + hn) * W_ + w0];
      n2 = *(const float4*)&gb[(size_t)(2 * H_ + hn) * W_ + w0];
      n3 = *(const float4*)&gb[(size_t)(3 * H_ + hn) * W_ + w0];
    }
    const float* pv = prevbuf[i & 1];
    float pm1 = (w0 > 0)      ? pv[w0 - 1] : 0.f;
    float p0 = pv[w0 + 0], p1 = pv[w0 + 1], p2 = pv[w0 + 2], p3 = pv[w0 + 3];
    float p4  = (w0 + 4 < W_) ? pv[w0 + 4] : 0.f;
    float c0 = fmaf(a0.x, xv.x, fmaf(a1.x, pm1, fmaf(a2.x, p0, a3.x * p1)));
    float c1 = fmaf(a0.y, xv.y, fmaf(a1.y, p0,  fmaf(a2.y, p1, a3.y * p2)));
    float c2 = fmaf(a0.z, xv.z, fmaf(a1.z, p1,  fmaf(a2.z, p2, a3.z * p3)));
    float c3 = fmaf(a0.w, xv.w, fmaf(a1.w, p2,  fmaf(a2.w, p3, a3.w * p4)));
    float* nv = prevbuf[(i + 1) & 1];
    nv[w0 + 0] = c0; nv[w0 + 1] = c1; nv[w0 + 2] = c2; nv[w0 + 3] = c3;
    *(float4*)&xo[(size_t)h * W_ + w0] = make_float4(c0, c1, c2, c3);
    __syncthreads();
    h = hn; xv = nx; a0 = n0; a1 = n1; a2 = n2; a3 = n3;
  }
}

// ---------------------------------------------------------------------------
// W sweep (DIR=+1 right, DIR=-1 left). One block per (b,c) plane, 256 threads
// (one per row h). Column-major global access is staged through LDS tiles
// filled by async global->LDS B128 copies, double-buffered across 16-column
// chunks. Output chunk staged in LDS, stored back with coalesced float4s.
// ---------------------------------------------------------------------------
template <int DIR>
__launch_bounds__(H_)
__global__ void nlf_sweepW(const float* in, const float* __restrict__ g, float* out) {
  extern __shared__ float smem[];
  float* tiles = smem;                   // 2 buffers * 5 tensors * TILEF
  float* outt  = smem + 2 * 5 * TILEF;   // TILEF
  float* prevb = outt + TILEF;           // 2 * H_ (double-buffered prev column)

  const int tid = threadIdx.x;           // 0..255 == row h
  const int bc  = blockIdx.x;
  const int b   = bc >> 5;
  const float* xin = in  + (size_t)bc * (H_ * W_);
  float*       xo  = out + (size_t)bc * (H_ * W_);
  const float* gb  = g   + (size_t)b  * (4 * H_ * W_);

  // Staging lane mapping: 4 lanes cover one 64B row segment (16 floats).
  const int r0 = tid >> 2;               // 0..63
  const int c0 = (tid & 3) * 4;          // 0,4,8,12

  prevb[tid] = 0.f;                      // prev column (buffer 0) = 0

  auto issue_chunk = [&](int w0, float* tb) {
#pragma unroll
    for (int p = 0; p < 4; ++p) {        // x tile: 4 B128s per thread
      const int r = r0 + 64 * p;
      async_load_b128(lds_off32(&tb[r * PITCH + c0]),
                      (uint64_t)(uintptr_t)&xin[(size_t)r * W_ + w0 + c0]);
    }
#pragma unroll
    for (int j = 0; j < 4; ++j) {        // g planes: 16 B128s per thread
      const float* gp = gb + (size_t)j * H_ * W_;
      float* tg = tb + (j + 1) * TILEF;
#pragma unroll
      for (int p = 0; p < 4; ++p) {
        const int r = r0 + 64 * p;
        async_load_b128(lds_off32(&tg[r * PITCH + c0]),
                        (uint64_t)(uintptr_t)&gp[(size_t)r * W_ + w0 + c0]);
      }
    }
  };

  // Prologue: fetch chunk 0 into buffer 0.
  issue_chunk((DIR > 0) ? 0 : (W_ - TW), tiles);

  for (int k = 0; k < NCHUNK; ++k) {
    const int buf = k & 1;
    if (k + 1 < NCHUNK) {
      const int w0n = (DIR > 0) ? (k + 1) * TW : (W_ - (k + 2) * TW);
      issue_chunk(w0n, tiles + (buf ^ 1) * (5 * TILEF));
      wait_async_20();                   // current chunk's 20 loads are done
    } else {
      wait_async_all();
    }
    __syncthreads();                     // tiles visible to all waves; outt reusable

    const float* tb = tiles + buf * (5 * TILEF);
    const int w0 = (DIR > 0) ? k * TW : (W_ - (k + 1) * TW);
    const int hh = tid;
#pragma unroll
    for (int j = 0; j < TW; ++j) {
      const int wc = (DIR > 0) ? j : (TW - 1 - j);
      const int s  = k * TW + j;         // global step index (parity continuous)
      const float* pv = prevb + (s & 1) * H_;
      float xv = tb[hh * PITCH + wc];
      float a0 = tb[1 * TILEF + hh * PITCH + wc];
      float a1 = tb[2 * TILEF + hh * PITCH + wc];
      float a2 = tb[3 * TILEF + hh * PITCH + wc];
      float a3 = tb[4 * TILEF + hh * PITCH + wc];
      float pl = (hh > 0)      ? pv[hh - 1] : 0.f;
      float pc = pv[hh];
      float pr = (hh < H_ - 1) ? pv[hh + 1] : 0.f;
      float cur = fmaf(a0, xv, fmaf(a1, pl, fmaf(a2, pc, a3 * pr)));
      prevb[((s + 1) & 1) * H_ + hh] = cur;
      outt[hh * PITCH + wc] = cur;
      __syncthreads();                   // single barrier per column step
    }
    // Coalesced write-back of the finished 16-column chunk.
#pragma unroll
    for (int p = 0; p < 4; ++p) {
      const int r = r0 + 64 * p;
      float4 v = *(const float4*)&outt[r * PITCH + c0];
      *(float4*)&xo[(size_t)r * W_ + w0 + c0] = v;
    }
  }
}

// ---------------------------------------------------------------------------
// Launcher: 4 dependent sweeps, fully in-place on d_out (sweep 1 writes every
// element of d_out from x before anything reads it; later sweeps read a global
// location only before overwriting it).
// ---------------------------------------------------------------------------
extern "C" void kernel_launch(void* const* d_in, const int* in_sizes, int n_in,
                              void* d_out, int out_size, void* d_ws, size_t ws_size,
                              hipStream_t stream) {
  (void)in_sizes; (void)n_in; (void)out_size; (void)d_ws; (void)ws_size;
  const float* x  = (const float*)d_in[0];
  const float* g0 = (const float*)d_in[1];
  const float* g1 = (const float*)d_in[2];
  const float* g2 = (const float*)d_in[3];
  const float* g3 = (const float*)d_in[4];
  float* out = (float*)d_out;

  dim3 gridP(B_ * C_);
  nlf_sweepH<+1><<<gridP, dim3(W_ / 4), 0, stream>>>(x,   g0, out);
  nlf_sweepH<-1><<<gridP, dim3(W_ / 4), 0, stream>>>(out, g1, out);

  constexpr size_t SH = (size_t)(2 * 5 * TILEF + TILEF + 2 * H_) * sizeof(float);
  nlf_sweepW<+1><<<gridP, dim3(H_), SH, stream>>>(out, g2, out);
  nlf_sweepW<-1><<<gridP, dim3(H_), SH, stream>>>(out, g3, out);
}